// RegionOfInterestPooling_10728828305831
// MI455X (gfx1250) — compile-verified
//
#include <hip/hip_runtime.h>
#include <stdint.h>

#define T_W 7
#define T_H 7
#define FMW 64
#define FMH 64
#define NCH 256
#define RPB 64              // regions per batch
#define MAXE 24             // max crop extent in fm coords (384/16)
#define CHUNKB 4096         // bytes per block-wide issue round (256 thr * 16B)

// Wait until this wave's outstanding async loads <= K (K block-uniform, 2..6).
__device__ __forceinline__ void wait_async_le(int K) {
    switch (K) {
        case 2:  asm volatile("s_wait_asynccnt 2" ::: "memory"); break;
        case 3:  asm volatile("s_wait_asynccnt 3" ::: "memory"); break;
        case 4:  asm volatile("s_wait_asynccnt 4" ::: "memory"); break;
        case 5:  asm volatile("s_wait_asynccnt 5" ::: "memory"); break;
        default: asm volatile("s_wait_asynccnt 6" ::: "memory"); break;
    }
}

__global__ __launch_bounds__(256) void roi_pool_async_kernel(
    const float* __restrict__ fm,      // (8, 64, 64, 256) f32
    const int*   __restrict__ regions, // (8, 64, 4) i32: x, y, w, h
    float*       __restrict__ out)     // (8, 64, 7, 7, 256) f32
{
    __shared__ float lds[2][MAXE * NCH];   // 2 x 24 KB double buffer

    const int blk = blockIdx.x;            // region index: bi*64 + r
    const int bi  = blk >> 6;              // blk / RPB
    const int c   = threadIdx.x;           // channel handled by this thread

    // ---- region geometry (matches reference integer math exactly) ----
    const int x = regions[blk * 4 + 0];
    const int y = regions[blk * 4 + 1];
    const int w = regions[blk * 4 + 2];
    const int h = regions[blk * 4 + 3];

    const int l  = x - w / 2,  t  = y - h / 2;
    const int rr = x + w / 2,  bb = y + h / 2;
    // image -> fm coords: floor(v/1024*64) == v>>4 (v >= 0 by construction)
    const int lf = l >> 4, rf = rr >> 4, tf = t >> 4, bf = bb >> 4;
    const int wdim = rf - lf;              // extent along fm axis0 (rows u)
    const int hdim = bf - tf;              // extent along fm axis1 (cols v)
    const int E  = min(wdim, hdim);        // valid extent on both axes (8..24)
    const int hs = max(hdim / T_H, 1);     // row step (single division, setup only)
    const int ws = max(wdim / T_W, 1);     // col step

    // chunks per row: cover E*1KB rounded up to whole 4KB chunks (block-uniform; 2..6)
    const int K = (E + 3) >> 2;

    // ---- async stage: one fm row segment (contiguous E KB) into LDS ----
    auto issue_row = [&](int i) {
        const int u = lf + i;
        const uint64_t gbase = (uint64_t)(uintptr_t)
            (fm + (((size_t)bi * FMW + (size_t)u) * FMH + (size_t)tf) * NCH);
        const uint32_t ldsbase = (uint32_t)(uintptr_t)(&lds[i & 1][0]);
        for (int k = 0; k < K; ++k) {
            const uint32_t voff  = (uint32_t)threadIdx.x * 16u + (uint32_t)(k * CHUNKB);
            const uint32_t laddr = ldsbase + voff;
            const uint64_t gaddr = gbase + voff;
            asm volatile("global_load_async_to_lds_b128 %0, %1, off"
                         :: "v"(laddr), "v"(gaddr) : "memory");
        }
    };

    float acc[T_W];
    #pragma unroll
    for (int tw = 0; tw < T_W; ++tw) acc[tw] = -__builtin_huge_valf();

    issue_row(0);

    const size_t obase = (size_t)blk * T_H * T_W * NCH;

    int th = 0;                 // current output row cell
    int boundary = hs;          // first row index belonging to the next cell

    for (int i = 0; i < E; ++i) {
        if (i + 1 < E) {
            issue_row(i + 1);   // prefetch next row into other buffer
            wait_async_le(K);   // in-order completion => row i fully in LDS
        } else {
            asm volatile("s_wait_asynccnt 0" ::: "memory");
        }
        __syncthreads();        // make all waves' async chunks of row i visible

        const float* rowbuf = &lds[i & 1][0];
        #pragma unroll
        for (int tw = 0; tw < T_W; ++tw) {
            const int cs = tw * ws;
            const int ce = (tw < T_W - 1) ? min((tw + 1) * ws, E) : E;
            float m = acc[tw];
            for (int j = cs; j < ce; ++j)
                m = fmaxf(m, rowbuf[j * NCH + c]);   // conflict-free ds_load_b32
            acc[tw] = m;
        }
        __syncthreads();        // all reads done before this buffer is re-filled (row i+2)

        // flush when the row-cell ends (boundaries are monotone; no division)
        const bool cell_end = (th < T_H - 1) ? ((i + 1) == boundary)
                                             : ((i + 1) == E);
        if (cell_end) {
            #pragma unroll
            for (int tw = 0; tw < T_W; ++tw) {
                out[obase + ((size_t)th * T_W + tw) * NCH + c] = acc[tw];
                acc[tw] = -__builtin_huge_valf();
            }
            ++th;
            boundary += hs;
        }
    }
}

extern "C" void kernel_launch(void* const* d_in, const int* in_sizes, int n_in,
                              void* d_out, int out_size, void* d_ws, size_t ws_size,
                              hipStream_t stream) {
    const float* fm      = (const float*)d_in[0];   // (8,64,64,256) f32
    const int*   regions = (const int*)d_in[1];     // (8,64,4) i32
    float*       out     = (float*)d_out;           // (8,64,7,7,256) f32

    dim3 grid(8 * RPB);   // one block per region
    dim3 block(256);      // 8 wave32s; thread == channel
    roi_pool_async_kernel<<<grid, block, 0, stream>>>(fm, regions, out);
}